// SpecCnn1d_v01_20529943675595
// MI455X (gfx1250) — compile-verified
//
#include <hip/hip_runtime.h>

typedef __attribute__((ext_vector_type(2))) float v2f;
typedef __attribute__((ext_vector_type(8))) float v8f;

// Problem constants (from reference): x (64, 16384) f32, kernel (128, 16) f32,
// stride 1 -> out (64, 16369, 128) f32 with ReLU.
constexpr int BATCH   = 64;
constexpr int LEN     = 16384;
constexpr int KW      = 16;
constexpr int NF      = 128;
constexpr int OUT_LEN = LEN - KW + 1;          // 16369
constexpr int OTILES  = (OUT_LEN + 15) / 16;   // 1024
constexpr int TILES   = BATCH * OTILES;        // 65536
constexpr int WAVES_PER_BLOCK = 8;             // 256 threads, wave32

__global__ __launch_bounds__(256) void
SpecCnn1d_wmma_f32_v2_kernel(const float* __restrict__ x,
                             const float* __restrict__ w,
                             float* __restrict__ out) {
    const int lane = threadIdx.x & 31;
    const int wave = threadIdx.x >> 5;
    const int tile = blockIdx.x * WAVES_PER_BLOCK + wave;
    if (tile >= TILES) return;                 // wave-uniform (never true: exact fit)

    const int b  = tile >> 10;                 // tile / OTILES
    const int ot = tile & (OTILES - 1);
    const int o0 = ot << 4;                    // first output position of tile

    const int m  = lane & 15;                  // row (M) / col (N) within 16x16
    const int hi = lane >> 4;                  // 0: K pair {0,1}; 1: K pair {2,3}

    // ---- A fragments: window rows. Clamp so the partial last tile never reads OOB.
    const int o  = o0 + m;
    const int oc = (o < OUT_LEN) ? o : (OUT_LEN - 1);
    const float* __restrict__ xrow = x + (size_t)b * LEN + oc;

    v2f a[4];
#pragma unroll
    for (int kb = 0; kb < 4; ++kb) {
        const int k = kb * 4 + hi * 2;
        a[kb].x = xrow[k];
        a[kb].y = xrow[k + 1];
    }

    // ---- Accumulate all 8 N-tiles (full F=128) before storing.
    v8f acc[8];
#pragma unroll
    for (int nt = 0; nt < 8; ++nt) {
        v8f c = {};
#pragma unroll
        for (int kb = 0; kb < 4; ++kb) {
            // B fragment: kernel[f = nt*16 + n][kb*4 + hi*2 .. +1]
            // even offset -> 8-byte aligned -> single global_load_b64
            const v2f bf = *(const v2f*)(w + (size_t)(nt * 16 + m) * KW
                                           + (kb * 4 + hi * 2));
            c = __builtin_amdgcn_wmma_f32_16x16x4_f32(
                    /*neg_a=*/false, a[kb],
                    /*neg_b=*/false, bf,
                    /*c_mod=*/(short)0, c,
                    /*reuse_a=*/false, /*reuse_b=*/false);
        }
        acc[nt] = c;
    }

    // ---- Store: one base pointer per lane, all 64 offsets are compile-time
    // constants (r*NF*4 + nt*16*4 bytes) -> global_store_b32 ... offset:imm.
    // Output is streamed (536 MB, write-once): non-temporal to spare L2.
    float* __restrict__ pout =
        out + ((size_t)b * OUT_LEN + o0 + hi * 8) * NF + m;

    if (o0 + 16 <= OUT_LEN) {                  // wave-uniform fast path (1023/1024)
#pragma unroll
        for (int nt = 0; nt < 8; ++nt) {
#pragma unroll
            for (int r = 0; r < 8; ++r) {
                __builtin_nontemporal_store(fmaxf(acc[nt][r], 0.0f),
                                            pout + r * NF + nt * 16);
            }
        }
    } else {                                   // tail tile: per-row guard
#pragma unroll
        for (int nt = 0; nt < 8; ++nt) {
#pragma unroll
            for (int r = 0; r < 8; ++r) {
                if (o0 + hi * 8 + r < OUT_LEN) {
                    __builtin_nontemporal_store(fmaxf(acc[nt][r], 0.0f),
                                                pout + r * NF + nt * 16);
                }
            }
        }
    }
}

extern "C" void kernel_launch(void* const* d_in, const int* in_sizes, int n_in,
                              void* d_out, int out_size, void* d_ws, size_t ws_size,
                              hipStream_t stream) {
    const float* x = (const float*)d_in[0];   // (64, 16384) f32
    const float* w = (const float*)d_in[1];   // (128, 16) f32
    float* out     = (float*)d_out;           // (64, 16369, 128) f32

    const int blocks = TILES / WAVES_PER_BLOCK; // 8192
    SpecCnn1d_wmma_f32_v2_kernel<<<blocks, 256, 0, stream>>>(x, w, out);
}